// SphericalSigmaBN_16243566313943
// MI455X (gfx1250) — compile-verified
//
#include <hip/hip_runtime.h>

// ---------------------------------------------------------------------------
// SphericalSigmaBN for MI455X (gfx1250, wave32)
// x: [8, 64, 256, 256, 2] f32.  Two streaming passes (reduce |x|, then scale).
// Wave-level reduction uses v_wmma_f32_16x16x4_f32 (B = ones) so the cross-
// lane sum runs on the matrix pipe at full f32 precision.
// ---------------------------------------------------------------------------

typedef float v2f __attribute__((ext_vector_type(2)));
typedef float v4f __attribute__((ext_vector_type(4)));   // native vec for b128
typedef float v8f __attribute__((ext_vector_type(8)));

#define B_DIM            8
#define CH               64
#define SLAB             131072            // 256*256*2 floats per (b, c)
#define NSLABS           (B_DIM * CH)      // 512
#define CHUNK            4096              // floats per block
#define CHUNKS_PER_SLAB  (SLAB / CHUNK)    // 32
#define PER_CH_COUNT     (B_DIM * SLAB)    // 1048576 elements per channel
#define NSEG             16

__device__ __constant__ int kSegId[CH] = {
    0,1,2,3,
    4,4,4, 5,5,5, 6,6,6, 7,7,7,
    8,8,8,8,8, 9,9,9,9,9, 10,10,10,10,10, 11,11,11,11,11,
    12,12,12,12,12,12,12, 13,13,13,13,13,13,13,
    14,14,14,14,14,14,14, 15,15,15,15,15,15,15};
__device__ __constant__ int kSegLo[NSEG]   = {0,1,2,3,4,7,10,13,16,21,26,31,36,43,50,57};
__device__ __constant__ int kSegSize[NSEG] = {1,1,1,1,3,3,3,3,5,5,5,5,7,7,7,7};

// --- kernel 0: zero the 64 f64 channel accumulators (ws is never re-poisoned)
__global__ void zero_ws_kernel(double* __restrict__ acc) {
    if (threadIdx.x < CH) acc[threadIdx.x] = 0.0;
}

// --- kernel 1: per-channel sum of |x|  (grid = [32 chunks, 512 slabs])
__global__ void reduce_abs_kernel(const float* __restrict__ x,
                                  double* __restrict__ acc) {
    const int  slab = blockIdx.y;
    const int  chan = slab & (CH - 1);           // slab = b*64 + c
    const long base = (long)slab * SLAB + (long)blockIdx.x * CHUNK;
    const v4f* __restrict__ xv = (const v4f*)(x + base);

    // 4 coalesced b128 loads per thread: lane-contiguous -> 512B per wave/instr
    float s = 0.0f;
#pragma unroll
    for (int i = 0; i < 4; ++i) {
        v4f v = xv[i * 256 + threadIdx.x];
        s += fabsf(v.x) + fabsf(v.y) + fabsf(v.z) + fabsf(v.w);
    }

    // Wave32 cross-lane sum on the matrix pipe:
    //   A[16x4]: lane L puts s into its slot (other slot 0), B = ones(4x16)
    //   => D[m][n] = s_m + s_(m+16)  for every column n.
    //   Lanes 0-15 hold D rows 0-7, lanes 16-31 hold rows 8-15 (8 VGPRs).
    v2f a; a[0] = s;    a[1] = 0.0f;
    v2f b; b[0] = 1.0f; b[1] = 1.0f;
    v8f c = {};
    v8f d = __builtin_amdgcn_wmma_f32_16x16x4_f32(
        /*neg_a=*/false, a, /*neg_b=*/false, b,
        /*c_mod=*/(short)0, c, /*reuse_a=*/false, /*reuse_b=*/false);
    float half_sum = d[0] + d[1] + d[2] + d[3] + d[4] + d[5] + d[6] + d[7];
    float wave_sum = half_sum + __shfl_xor(half_sum, 16, 32);  // rows 0-7 + 8-15

    __shared__ float wsum[8];
    const int lane = threadIdx.x & 31;
    const int wave = threadIdx.x >> 5;
    if (lane == 0) wsum[wave] = wave_sum;
    __syncthreads();

    if (threadIdx.x == 0) {
        float bs = 0.0f;
#pragma unroll
        for (int w = 0; w < 8; ++w) bs += wsum[w];
        atomicAdd(&acc[chan], (double)bs);      // global_atomic_add_f64
    }
}

// --- kernel 2: segment means -> momentum blend -> per-channel scale (1 block)
__global__ void finalize_kernel(const double* __restrict__ acc,
                                const float* __restrict__ sigma2,
                                const int* __restrict__ update,
                                float* __restrict__ chan_scale) {
    __shared__ float segscale[NSEG];
    const int t = threadIdx.x;
    if (t < NSEG) {
        const int lo = kSegLo[t];
        const int n  = kSegSize[t];
        double sum = 0.0;
        for (int cc = lo; cc < lo + n; ++cc)
            sum += acc[cc] * (1.0 / (double)PER_CH_COUNT);   // per-channel mean
        float s2  = (float)(sum / (double)n);                // segment mean
        float sig = sigma2[t];
        if (update[0]) sig = (sig == 0.0f) ? s2 : sig * 0.9f + s2 * 0.1f;
        segscale[t] = 1.0f / (sqrtf(sig) + 1e-5f);
    }
    __syncthreads();
    if (t < CH) chan_scale[t] = segscale[kSegId[t]];
}

// --- kernel 3: out = x * scale[channel]  (grid = [32 chunks, 512 slabs])
__global__ void scale_kernel(const float* __restrict__ x,
                             const float* __restrict__ chan_scale,
                             float* __restrict__ out) {
    const int  slab = blockIdx.y;
    const int  chan = slab & (CH - 1);
    const float sc  = chan_scale[chan];          // uniform -> scalar load
    const long base = (long)slab * SLAB + (long)blockIdx.x * CHUNK;
    const v4f* __restrict__ xv = (const v4f*)(x + base);
    v4f* __restrict__ ov = (v4f*)(out + base);
#pragma unroll
    for (int i = 0; i < 4; ++i) {
        v4f v = xv[i * 256 + threadIdx.x];
        v *= sc;                                               // broadcast mul
        __builtin_nontemporal_store(v, &ov[i * 256 + threadIdx.x]);  // NT b128
    }
}

extern "C" void kernel_launch(void* const* d_in, const int* in_sizes, int n_in,
                              void* d_out, int out_size, void* d_ws, size_t ws_size,
                              hipStream_t stream) {
    const float* x      = (const float*)d_in[0];   // [8,64,256,256,2] f32
    const float* sigma2 = (const float*)d_in[1];   // [16] f32
    const int*   update = (const int*)d_in[2];     // scalar int
    float*       out    = (float*)d_out;

    double* acc        = (double*)d_ws;                               // 64 f64
    float*  chan_scale = (float*)((char*)d_ws + CH * sizeof(double)); // 64 f32

    zero_ws_kernel<<<1, 64, 0, stream>>>(acc);

    dim3 grid(CHUNKS_PER_SLAB, NSLABS);   // 32 x 512 = 16384 blocks
    reduce_abs_kernel<<<grid, 256, 0, stream>>>(x, acc);
    finalize_kernel<<<1, 64, 0, stream>>>(acc, sigma2, update, chan_scale);
    scale_kernel<<<grid, 256, 0, stream>>>(x, chan_scale, out);

    (void)in_sizes; (void)n_in; (void)out_size; (void)ws_size;
}